// ConvTrBlock2D_12790412608006
// MI455X (gfx1250) — compile-verified
//
#include <hip/hip_runtime.h>
#include <hip/hip_bf16.h>

typedef float v2f __attribute__((ext_vector_type(2)));
typedef float v4f __attribute__((ext_vector_type(4)));
typedef float v8f __attribute__((ext_vector_type(8)));

#define CIN   128
#define COUT  64
#define KVOL  4
#define BN_EPS 1e-5f

// ---- ws header layout (bytes) ----
// [0,16)    counts[4]   (uint)
// [16,32)   offsets[4]
// [32,48)   cursors[4]
// [64,320)  gsum[64]    (float)
// [320,576) gsq[64]     (float)
// [1024,..) perm[M]     (int)

__global__ void zero_header_kernel(unsigned int* w) {
    w[threadIdx.x] = 0u;            // 256 uints = first 1KB (counts..gsq)
}

__global__ void count_kernel(const int* __restrict__ kidx, unsigned int* __restrict__ counts, int M) {
    int i = blockIdx.x * blockDim.x + threadIdx.x;
    if (i < M) atomicAdd(&counts[kidx[i]], 1u);
}

__global__ void offsets_kernel(unsigned int* w) {
    unsigned int o = 0;
    for (int k = 0; k < KVOL; ++k) {
        w[4 + k] = o;               // offsets
        w[8 + k] = o;               // cursors (init = offsets)
        o += w[k];
    }
}

__global__ void scatter_kernel(const int* __restrict__ kidx, unsigned int* __restrict__ cursors,
                               int* __restrict__ perm, int M) {
    int i = blockIdx.x * blockDim.x + threadIdx.x;
    if (i < M) {
        unsigned int p = atomicAdd(&cursors[kidx[i]], 1u);
        perm[p] = i;
    }
}

// 8 waves per 256-thread block; each wave owns a 16x64 tile computed with
// V_WMMA_F32_16X16X4_F32. B fragments are staged in LDS with K-pairs
// interleaved so each (K=2p,K=2p+1,n) fragment is one ds_load_b64.
// perm is bucket-sorted by kidx, so a block's 128 rows span a contiguous
// k-range of at most 2 weight matrices (bucket size >> 128) -> 64KB LDS.
__global__ __launch_bounds__(256) void gemm_wmma_kernel(
    const float* __restrict__ feats, const float* __restrict__ weight,
    const int* __restrict__ in_idx, const int* __restrict__ kidx,
    const int* __restrict__ perm, float* __restrict__ out,
    float* __restrict__ gsum, float* __restrict__ gsq, int ntiles, int M)
{
    extern __shared__ float smem[];
    float* lds_b = smem;                       // [2][64 pairs][64 n][2] floats
    float* s_sum = smem + 2 * 64 * 64 * 2;     // [64]
    float* s_sq  = s_sum + COUT;               // [64]

    int t = threadIdx.x;
    if (t < COUT) { s_sum[t] = 0.0f; s_sq[t] = 0.0f; }

    // k-range of this block's rows (contiguous because perm is bucket-sorted)
    int r0 = blockIdx.x * 128;
    int rl = min(r0 + 127, M - 1);
    int kf = kidx[perm[r0]];
    int kl = kidx[perm[rl]];
    int nk = min(kl - kf + 1, 2);

    // Stage B for k in [kf, kf+nk): interleave rows 2p and 2p+1 so that
    // LDS holds (K=2p,n),(K=2p+1,n) adjacent -> b64 fragment loads.
    for (int ks = 0; ks < nk; ++ks) {
        const float* __restrict__ wk = weight + (long)(kf + ks) * (CIN * COUT);
        for (int c = t; c < 1024; c += 256) {       // 64 pairs x 16 chunks of 4 n
            int p  = c >> 4;
            int nc = (c & 15) << 2;
            v4f ra = *(const v4f*)(wk + (long)(2 * p) * COUT + nc);
            v4f rb = *(const v4f*)(wk + (long)(2 * p + 1) * COUT + nc);
            float* d = lds_b + ((ks * 64 + p) * 64 + nc) * 2;
            v4f lo = {ra.x, rb.x, ra.y, rb.y};
            v4f hi = {ra.z, rb.z, ra.w, rb.w};
            *(v4f*)d       = lo;
            *(v4f*)(d + 4) = hi;
        }
    }
    __syncthreads();

    int wave = t >> 5;
    int lane = t & 31;
    int half = lane >> 4;           // K-half selector in A/B fragment layout
    int l16  = lane & 15;           // M index (A) / N index (B,C)
    int tile = blockIdx.x * 8 + wave;

    if (tile < ntiles) {
        int m  = tile * 16 + l16;
        int pm = perm[m];                        // original output row
        int fr = in_idx[pm];
        int kr = kidx[pm];                       // weight id for this row
        const float* __restrict__ arow = feats + (long)fr * CIN;

        v8f c0 = {}, c1 = {}, c2 = {}, c3 = {};

        for (int k = kf; k <= kl; ++k) {
            if (!__any(kr == k)) continue;       // wave-uniform; EXEC stays full
            float amask = (kr == k) ? 1.0f : 0.0f;
            const float* __restrict__ bb = lds_b + min(k - kf, 1) * (64 * 64 * 2);
            for (int kk = 0; kk < CIN; kk += 4) {
                int ka = kk + half * 2;
                int p  = ka >> 1;                // staged pair index
                v2f a;
                a.x = amask * arow[ka];
                a.y = amask * arow[ka + 1];
                const float* br = bb + (p * 64 + l16) * 2;
                v2f b0 = *(const v2f*)(br);       // n-tile 0
                v2f b1 = *(const v2f*)(br + 32);  // n-tile 1 (+16 cols)
                v2f b2 = *(const v2f*)(br + 64);  // n-tile 2
                v2f b3 = *(const v2f*)(br + 96);  // n-tile 3
                c0 = __builtin_amdgcn_wmma_f32_16x16x4_f32(false, a, false, b0, (short)0, c0, false, false);
                c1 = __builtin_amdgcn_wmma_f32_16x16x4_f32(false, a, false, b1, (short)0, c1, false, false);
                c2 = __builtin_amdgcn_wmma_f32_16x16x4_f32(false, a, false, b2, (short)0, c2, false, false);
                c3 = __builtin_amdgcn_wmma_f32_16x16x4_f32(false, a, false, b3, (short)0, c3, false, false);
            }
        }

        // C/D layout: VGPR v, lanes 0-15 -> (M=v, N=lane); lanes 16-31 -> (M=v+8, N=lane-16)
        auto store_tile = [&](const v8f& c, int nt) {
            float s = 0.0f, q = 0.0f;
            int col = nt * 16 + l16;
            #pragma unroll
            for (int v = 0; v < 8; ++v) {
                int prow = __shfl(pm, (half << 3) + v);   // perm of row (half*8+v)
                float val = c[v];
                out[(long)prow * COUT + col] = val;
                s += val;
                q += val * val;
            }
            atomicAdd(&s_sum[col], s);
            atomicAdd(&s_sq[col], q);
        };
        store_tile(c0, 0);
        store_tile(c1, 1);
        store_tile(c2, 2);
        store_tile(c3, 3);
    }

    __syncthreads();
    if (t < COUT) {
        atomicAdd(&gsum[t], s_sum[t]);
        atomicAdd(&gsq[t],  s_sq[t]);
    }
}

__global__ void bn_relu_kernel(float* __restrict__ out,
                               const float* __restrict__ gsum, const float* __restrict__ gsq,
                               const float* __restrict__ gamma, const float* __restrict__ beta,
                               int total, float invM)
{
    int i = blockIdx.x * blockDim.x + threadIdx.x;
    if (i >= total) return;
    int col = i & (COUT - 1);
    float mean = gsum[col] * invM;
    float var  = gsq[col] * invM - mean * mean;   // biased variance, as in reference
    float scale = gamma[col] * rsqrtf(var + BN_EPS);
    float y = (out[i] - mean) * scale + beta[col];
    out[i] = fmaxf(y, 0.0f);
}

extern "C" void kernel_launch(void* const* d_in, const int* in_sizes, int n_in,
                              void* d_out, int out_size, void* d_ws, size_t ws_size,
                              hipStream_t stream) {
    const float* feats  = (const float*)d_in[0];   // [N_IN, 128]
    const float* weight = (const float*)d_in[1];   // [4, 128, 64]
    const float* gamma  = (const float*)d_in[2];   // [64]
    const float* beta   = (const float*)d_in[3];   // [64]
    const int*   in_idx = (const int*)d_in[4];     // [M]
    const int*   kidx   = (const int*)d_in[5];     // [M]
    float* out = (float*)d_out;                    // [M, 64]

    int M = in_sizes[4];
    int ntiles = M / 16;                           // M = 600000 -> 37500 tiles

    unsigned int* hdr     = (unsigned int*)d_ws;
    unsigned int* counts  = hdr;
    unsigned int* cursors = hdr + 8;
    float* gsum = (float*)((char*)d_ws + 64);
    float* gsq  = (float*)((char*)d_ws + 320);
    int*   perm = (int*)((char*)d_ws + 1024);

    zero_header_kernel<<<1, 256, 0, stream>>>(hdr);
    count_kernel<<<(M + 255) / 256, 256, 0, stream>>>(kidx, counts, M);
    offsets_kernel<<<1, 1, 0, stream>>>(hdr);
    scatter_kernel<<<(M + 255) / 256, 256, 0, stream>>>(kidx, cursors, perm, M);

    int gblocks = (ntiles + 7) / 8;
    size_t shmem = (size_t)(2 * 64 * 64 * 2 + 2 * COUT) * sizeof(float);  // 66048 B
    gemm_wmma_kernel<<<gblocks, 256, shmem, stream>>>(
        feats, weight, in_idx, kidx, perm, out, gsum, gsq, ntiles, M);

    int total = M * COUT;
    bn_relu_kernel<<<(total + 255) / 256, 256, 0, stream>>>(
        out, gsum, gsq, gamma, beta, total, 1.0f / (float)M);
}